// QuantumConvolutionLayer_61942018343742
// MI455X (gfx1250) — compile-verified
//
#include <hip/hip_runtime.h>
#include <cstdint>
#include <cstddef>

#define H_IN 512
#define W_IN 512
#define HO   256
#define WO   256
#define NB   64
#define ROWS 8

// ---------------------------------------------------------------------------
// Main kernel FIRST in the module so the disasm snippet shows it.
//
// Memory-bound (~64 MB total traffic -> ~2.7us at 23.3 TB/s): each block
// stages 8 input rows (16 KB) into LDS via CDNA5 async global->LDS B128 DMA
// (ASYNCcnt path), then 256 threads emit 8x256 pixels. LDS read pattern
// (stride-2 words) hits 64 distinct banks: conflict-free in a wave32.
// Per pixel: z_q = u0^T B_q u1, u = (1, cos(pi*phi), sin(pi*phi)); the 18
// B-coefficients are produced by the tiny precompute kernel below.
// ---------------------------------------------------------------------------
__global__ __launch_bounds__(256)
void qconv_main(const float* __restrict__ img,
                const float* __restrict__ Bc,
                float* __restrict__ out) {
  __shared__ float tile[ROWS * W_IN];   // 16 KB of 320 KB WGP LDS

  const int tid = threadIdx.x;          // 0..255 (8 wave32)
  const int b   = blockIdx.x >> 5;      // batch
  const int i0  = (blockIdx.x & 31) * ROWS;

  // ---- async stage: rows ja[i0..i0+7] (ja = min(2i+1, 510)), 2 KB each ----
  {
    const int chunk = tid & 127;        // 16-byte chunk within a row
    const int rhalf = tid >> 7;         // two rows per issue round
#pragma unroll
    for (int k = 0; k < ROWS; k += 2) {
      const int r = k + rhalf;
      int grow = 2 * (i0 + r) + 1;
      if (grow > H_IN - 2) grow = H_IN - 2;
      const float* gsrc = img + ((size_t)b * H_IN + grow) * W_IN + (size_t)chunk * 4;
      unsigned long long gaddr = (unsigned long long)(size_t)gsrc;
      unsigned lds_off = (unsigned)(size_t)(&tile[r * W_IN + chunk * 4]);
      asm volatile("global_load_async_to_lds_b128 %0, %1, off"
                   :: "v"(lds_off), "v"(gaddr) : "memory");
    }
  }
#if __has_builtin(__builtin_amdgcn_s_wait_asynccnt)
  __builtin_amdgcn_s_wait_asynccnt(0);
#else
  asm volatile("s_wait_asynccnt 0" ::: "memory");
#endif
  __syncthreads();

  // 18 uniform coefficients -> scalar loads, L2-resident
  float B0[9], B1[9];
#pragma unroll
  for (int t = 0; t < 9; ++t) { B0[t] = Bc[t]; B1[t] = Bc[9 + t]; }

  const int j = tid;                          // output column
  int cw = 2 * j + 1;                         // kb[j] = min(2j+1, 510)
  if (cw > W_IN - 2) cw = W_IN - 2;
  const size_t obase = (size_t)b * 2u * (HO * WO) + (size_t)i0 * WO + (size_t)j;
  const float PI_F = 3.14159265358979323846f;

#pragma unroll
  for (int r = 0; r < ROWS; ++r) {
    const float phi0 = tile[r * W_IN + cw];
    const float phi1 = tile[r * W_IN + cw + 1];
    float S0, C0, S1, C1;
    __sincosf(phi0 * PI_F, &S0, &C0);         // v_sin/v_cos (arg/2pi = phi/2)
    __sincosf(phi1 * PI_F, &S1, &C1);
    const float w0 = B0[0] + B0[1] * C1 + B0[2] * S1;
    const float w1 = B0[3] + B0[4] * C1 + B0[5] * S1;
    const float w2 = B0[6] + B0[7] * C1 + B0[8] * S1;
    const float v0 = B1[0] + B1[1] * C1 + B1[2] * S1;
    const float v1 = B1[3] + B1[4] * C1 + B1[5] * S1;
    const float v2 = B1[6] + B1[7] * C1 + B1[8] * S1;
    const float z0 = w0 + C0 * w1 + S0 * w2;
    const float z1 = v0 + C0 * v1 + S0 * v2;
    out[obase + (size_t)r * WO]             = z0;   // plane q=0
    out[obase + (size_t)r * WO + HO * WO]   = z1;   // plane q=1
  }
}

// ---------------------------------------------------------------------------
// numpy legacy RandomState(42) replication (MT19937, init_genrand seeding),
// LAZY variant: the circuit consumes < 128 draws and every consumed index is
// < 227, so twist output i depends only on SEEDED words mt[i], mt[i+1],
// mt[i+397] (no in-place-update hazard). We seed 526 words and temper on
// demand -> ~5x fewer serial scalar ops than init+full-twist.
//   np.random.random()                -> rand01 (2x 32-bit draws)
//   np.random.choice(2,1,repl=False)  -> permutation(2): 1x 32-bit draw
//   np.random.choice(['RX','RY','RZ'])-> legacy randint(0,3): masked-reject
//                                        on 64-bit draws (2x 32-bit per try)
// ---------------------------------------------------------------------------
#define MT_DRAWS_MAX 128
#define MT_STATE     (MT_DRAWS_MAX + 398)   // 526

struct MT { unsigned mt[MT_STATE]; int idx; };

__device__ inline void mt_seed(MT& s, unsigned v) {
  s.mt[0] = v;
  for (int i = 1; i < MT_STATE; ++i)
    s.mt[i] = 1812433253u * (s.mt[i - 1] ^ (s.mt[i - 1] >> 30)) + (unsigned)i;
  s.idx = 0;
}
__device__ inline unsigned mt_next(MT& s) {
  int i = s.idx;
  if (i < MT_DRAWS_MAX) s.idx = i + 1;        // clamp (never hit in practice)
  unsigned y = (s.mt[i] & 0x80000000u) | (s.mt[i + 1] & 0x7fffffffu);
  unsigned v = s.mt[i + 397] ^ (y >> 1);
  if (y & 1u) v ^= 2567483615u;
  v ^= v >> 11;
  v ^= (v << 7)  & 2636928640u;
  v ^= (v << 15) & 4022730752u;
  v ^= v >> 18;
  return v;
}
__device__ inline double mt_rand01(MT& s) {
  unsigned a = mt_next(s) >> 5, b = mt_next(s) >> 6;
  return (a * 67108864.0 + b) / 9007199254740992.0;
}
__device__ inline int mt_interval1(MT& s) { return (int)(mt_next(s) & 1u); }
__device__ inline int mt_randint3(MT& s) {
  for (;;) {
    unsigned hi = mt_next(s); (void)hi;       // high word of 64-bit draw
    unsigned lo = mt_next(s);
    unsigned v = lo & 3u;                     // (hi<<32|lo) & mask(=3)
    if (v <= 2u) return (int)v;               // reject 3
  }
}

// ---------------------------------------------------------------------------
// Precompute kernel: build composite 4x4 complex unitary M = G_n ... G_1,
// then reduce the two observables to 3x3 bilinear forms B_q (18 floats).
// ---------------------------------------------------------------------------
__global__ void qconv_coeffs(const float* __restrict__ wts,
                             float* __restrict__ Bout) {
  __shared__ MT s;
  if (threadIdx.x != 0) return;
  mt_seed(s, 42u);

  float Mr[4][4] = {{1,0,0,0},{0,1,0,0},{0,0,1,0},{0,0,0,1}};
  float Mi[4][4] = {{0,0,0,0},{0,0,0,0},{0,0,0,0},{0,0,0,0}};

  int guard = 0;
  for (int l = 0; l < 2; ++l) {
    int i = 0;
    while (i < 2 && guard < 48) {
      ++guard;
      float Gr[4][4], Gi[4][4];
      for (int a = 0; a < 4; ++a)
        for (int b = 0; b < 4; ++b) { Gr[a][b] = 0.f; Gi[a][b] = 0.f; }

      if (mt_rand01(s) > 0.3) {                    // rotation gate
        int wq = 1 - mt_interval1(s);              // choice(2,1,replace=False)
        int g  = mt_randint3(s);                   // 0=RX 1=RY 2=RZ
        float th = wts[l * 2 + i];
        float ch = cosf(0.5f * th), sh = sinf(0.5f * th);
        float Ur[2][2] = {{ch, 0.f}, {0.f, ch}};
        float Ui[2][2] = {{0.f, 0.f}, {0.f, 0.f}};
        if (g == 0)      { Ui[0][1] = -sh; Ui[1][0] = -sh; }          // RX
        else if (g == 1) { Ur[0][1] = -sh; Ur[1][0] =  sh; }          // RY
        else             { Ui[0][0] = -sh; Ui[1][1] =  sh; }          // RZ
        for (int a = 0; a < 2; ++a)
          for (int b = 0; b < 2; ++b)
            for (int c = 0; c < 2; ++c)
              for (int d = 0; d < 2; ++d) {
                if (wq == 0) { if (c == d) { Gr[2*a+c][2*b+d] = Ur[a][b]; Gi[2*a+c][2*b+d] = Ui[a][b]; } }
                else         { if (a == b) { Gr[2*a+c][2*b+d] = Ur[c][d]; Gi[2*a+c][2*b+d] = Ui[c][d]; } }
              }
        ++i;
      } else {                                     // CNOT
        int j = mt_interval1(s);                   // permutation(2)
        if (j == 1) { Gr[0][0]=1.f; Gr[1][1]=1.f; Gr[2][3]=1.f; Gr[3][2]=1.f; } // (0,1)
        else        { Gr[0][0]=1.f; Gr[1][3]=1.f; Gr[2][2]=1.f; Gr[3][1]=1.f; } // (1,0)
      }

      // M = G * M   (psi' = G psi column form of psi @ G.T)
      float Nr[4][4], Ni[4][4];
      for (int r = 0; r < 4; ++r)
        for (int c = 0; c < 4; ++c) {
          float ar = 0.f, ai = 0.f;
          for (int k = 0; k < 4; ++k) {
            ar += Gr[r][k] * Mr[k][c] - Gi[r][k] * Mi[k][c];
            ai += Gr[r][k] * Mi[k][c] + Gi[r][k] * Mr[k][c];
          }
          Nr[r][c] = ar; Ni[r][c] = ai;
        }
      for (int r = 0; r < 4; ++r)
        for (int c = 0; c < 4; ++c) { Mr[r][c] = Nr[r][c]; Mi[r][c] = Ni[r][c]; }
    }
  }

  // z_q = psi^T A_q psi; psi = (c0,s0)(x)(c1,s1) separable & real
  //  -> z_q = u0^T B_q u1 with u = (1, cos(pi phi), sin(pi phi))
  const int   Pr[3][2][2] = {{{0,0},{0,0}}, {{1,1},{1,1}}, {{0,1},{1,0}}};
  const int   Pc[3]       = {1, 1, 2};
  const float F[3][3]     = {{0.5f,0.5f,0.f}, {0.5f,-0.5f,0.f}, {0.f,0.f,0.5f}};

  for (int q = 0; q < 2; ++q) {
    float A[4][4];
    for (int jj = 0; jj < 4; ++jj)
      for (int kk = 0; kk < 4; ++kk) {
        float acc = 0.f;
        for (int i = 0; i < 4; ++i) {
          int bit = (q == 0) ? ((i >> 1) & 1) : (i & 1);
          float sg = bit ? -1.f : 1.f;
          acc += sg * (Mr[i][jj] * Mr[i][kk] + Mi[i][jj] * Mi[i][kk]);
        }
        A[jj][kk] = acc;
      }
    float K[3][3];
    for (int m = 0; m < 3; ++m)
      for (int n = 0; n < 3; ++n) {
        float acc = 0.f;
        for (int p = 0; p < Pc[m]; ++p)
          for (int t = 0; t < Pc[n]; ++t) {
            int a = Pr[m][p][0], b = Pr[m][p][1];
            int c = Pr[n][t][0], d = Pr[n][t][1];
            acc += A[2*a + c][2*b + d];
          }
        K[m][n] = acc;
      }
    for (int r = 0; r < 3; ++r)
      for (int c = 0; c < 3; ++c) {
        float acc = 0.f;
        for (int m = 0; m < 3; ++m)
          for (int n = 0; n < 3; ++n)
            acc += K[m][n] * F[m][r] * F[n][c];
        Bout[q * 9 + r * 3 + c] = acc;
      }
  }
}

// ---------------------------------------------------------------------------
extern "C" void kernel_launch(void* const* d_in, const int* in_sizes, int n_in,
                              void* d_out, int out_size, void* d_ws, size_t ws_size,
                              hipStream_t stream) {
  (void)in_sizes; (void)n_in; (void)out_size; (void)ws_size;
  const float* x    = (const float*)d_in[0];   // (64,1,512,512) f32
  const float* wts  = (const float*)d_in[1];   // (2,2) f32
  float*       out  = (float*)d_out;           // (64,2,256,256) f32
  float*       coef = (float*)d_ws;            // 18 floats scratch

  qconv_coeffs<<<dim3(1), dim3(32), 0, stream>>>(wts, coef);
  qconv_main<<<dim3(NB * (HO / ROWS)), dim3(256), 0, stream>>>(x, coef, out);
}